// Fourier3dMesh_74844100100231
// MI455X (gfx1250) — compile-verified
//
#include <hip/hip_runtime.h>
#include <hip/hip_bf16.h>
#include <math.h>

// ---------------------------------------------------------------------------
// Problem constants (match reference)
// ---------------------------------------------------------------------------
#define N0 32
#define N1 32
#define N2 32
#define NGRID (N0 * N1 * N2)   // 32768
#define NV 258
#define NF 512
#define NTILES (NF / 16)       // 32 face tiles of 16
#define THRESH 0.01f

typedef __attribute__((ext_vector_type(2))) float v2f;
typedef __attribute__((ext_vector_type(8))) float v8f;

// Workspace layout (floats):
//   [0, 512)            trisar
//   [512, 512 + 6144)   Bpack: [tile(32)][vert(3)][lane(32)][2]
//                       lanes 0-15:  {x, y}   (K0, K1)
//                       lanes 16-31: {z, 0}   (K2, K3)
#define WS_TRISAR 0
#define WS_BPACK  512

__device__ __forceinline__ float fast_rcp(float x) {
  return __builtin_amdgcn_rcpf(x);
}

// ---------------------------------------------------------------------------
// Kernel 1: per-face areas (Heron), global normalization, WMMA-ready B pack.
// One block of 512 threads == one thread per face.
// ---------------------------------------------------------------------------
__global__ void __launch_bounds__(NF)
mesh_ft_precompute(const float* __restrict__ Verts,
                   const int*   __restrict__ Faces,
                   const float* __restrict__ coeff,
                   float*       __restrict__ ws) {
  __shared__ float red[NF];
  const int f = threadIdx.x;

  const int ia = Faces[3 * f + 0];
  const int ib = Faces[3 * f + 1];
  const int ic = Faces[3 * f + 2];

  // box_min == 0 for this problem, so V == Verts.
  const float ax = Verts[3 * ia + 0], ay = Verts[3 * ia + 1], az = Verts[3 * ia + 2];
  const float bx = Verts[3 * ib + 0], by = Verts[3 * ib + 1], bz = Verts[3 * ib + 2];
  const float cx = Verts[3 * ic + 0], cy = Verts[3 * ic + 1], cz = Verts[3 * ic + 2];

  // sides = pos - pos[[2,0,1]] -> [a-c, b-a, c-b]
  const float s0x = ax - cx, s0y = ay - cy, s0z = az - cz;
  const float s1x = bx - ax, s1y = by - ay, s1z = bz - az;
  const float s2x = cx - bx, s2y = cy - by, s2z = cz - bz;
  const float L0 = sqrtf(s0x * s0x + s0y * s0y + s0z * s0z);
  const float L1 = sqrtf(s1x * s1x + s1y * s1y + s1z * s1z);
  const float L2 = sqrtf(s2x * s2x + s2y * s2y + s2z * s2z);
  const float s  = 0.5f * (L0 + L1 + L2);
  const float area = coeff[f] * sqrtf(s * (s - L0) * (s - L1) * (s - L2));

  red[f] = area;
  __syncthreads();
  for (int off = NF / 2; off > 0; off >>= 1) {
    if (f < off) red[f] += red[f + off];
    __syncthreads();
  }
  const float meshar = red[0];

  ws[WS_TRISAR + f] = area * (2.0f / meshar);

  // WMMA B-operand pack: per tile of 16 faces, per vertex, per lane a v2f.
  //   lane (f&15)      -> {x, y}  (K rows 0,1)
  //   lane (f&15)+16   -> {z, 0}  (K rows 2,3)
  float* Bp = ws + WS_BPACK;
  const int tile = f >> 4;
  const int mm   = f & 15;
  const float vx[3] = {ax, bx, cx};
  const float vy[3] = {ay, by, cy};
  const float vz[3] = {az, bz, cz};
#pragma unroll
  for (int v = 0; v < 3; ++v) {
    const int base = ((tile * 3 + v) * 32);
    Bp[(base + mm) * 2 + 0]      = vx[v];
    Bp[(base + mm) * 2 + 1]      = vy[v];
    Bp[(base + mm + 16) * 2 + 0] = vz[v];
    Bp[(base + mm + 16) * 2 + 1] = 0.0f;
  }
}

// ---------------------------------------------------------------------------
// Kernel 2: main. Each wave owns 16 grid points; loops over 32 face tiles.
// xia/xib/xic tiles via V_WMMA_F32_16X16X4_F32 (K holds x,y,z,0).
// ---------------------------------------------------------------------------
__global__ void __launch_bounds__(256)
mesh_ft_main(const float* __restrict__ xi0,
             const float* __restrict__ xi1,
             const float* __restrict__ xi2,
             const float* __restrict__ OTF,
             const float* __restrict__ ws,
             float*       __restrict__ out) {
  const float SEPS = 3.4526699e-4f;  // sqrtf(1.19209e-07f)

  const float* trisar = ws + WS_TRISAR;
  const v2f*   Bp     = (const v2f*)(ws + WS_BPACK);

  const int lane = threadIdx.x & 31;
  const int m    = lane & 15;        // face column within tile / A-matrix row
  const int hi   = lane >> 4;        // half-wave selector (K=2,3 resp. M+8)
  const int wid  = blockIdx.x * (blockDim.x >> 5) + (threadIdx.x >> 5);
  const int gbase = wid * 16;        // 16 grid points per wave

  // Grid-point coordinates for A-matrix row m = lane%16.
  const int g  = gbase + m;
  const int gi = g >> 10;            // / (N1*N2)
  const int gj = (g >> 5) & 31;
  const int gk = g & 31;
  const float c0 = xi0[gi];
  const float c1 = xi1[gj];
  const float c2 = xi2[gk];

  // A (16x4 f32): lane<16 -> {K0=c0, K1=c1}; lane>=16 -> {K2=c2, K3=0}
  v2f A;
  A.x = hi ? c2 : c0;
  A.y = hi ? 0.0f : c1;

  float accRe[8], accIm[8];
#pragma unroll
  for (int r = 0; r < 8; ++r) { accRe[r] = 0.0f; accIm[r] = 0.0f; }

#pragma unroll 1
  for (int tile = 0; tile < NTILES; ++tile) {
    const float t = trisar[tile * 16 + m];  // face weight (same for both halves)

    // WMMA-ready B operands: one coalesced b64 load per vertex, no selects.
    const v2f Ba = Bp[(tile * 3 + 0) * 32 + lane];
    const v2f Bb = Bp[(tile * 3 + 1) * 32 + lane];
    const v2f Bc = Bp[(tile * 3 + 2) * 32 + lane];

    v8f cz = {};
    // xia/xib/xic 16x16 tiles: D = A x B  (K=4, f32)
    v8f Da = __builtin_amdgcn_wmma_f32_16x16x4_f32(false, A, false, Ba,
                                                   (short)0, cz, false, false);
    v8f Db = __builtin_amdgcn_wmma_f32_16x16x4_f32(false, A, false, Bb,
                                                   (short)0, cz, false, false);
    v8f Dc = __builtin_amdgcn_wmma_f32_16x16x4_f32(false, A, false, Bc,
                                                   (short)0, cz, false, false);

#pragma unroll
    for (int r = 0; r < 8; ++r) {
      const float xa = Da[r];
      const float xb = Db[r];
      const float xc = Dc[r];

      const float xiab = xb - xa;
      const float xibc = xc - xb;
      const float xica = xa - xc;
      const float aab = fabsf(xiab);
      const float abc = fabsf(xibc);
      const float aca = fabsf(xica);

      const bool ab_lt = aab < SEPS;
      const bool bc_lt = abc < SEPS;
      const bool ca_lt = aca < SEPS;
      const bool R1 = (ab_lt && bc_lt) || (bc_lt && ca_lt) || (ca_lt && ab_lt);
      const bool R2 = (aab <= SEPS) && !R1;
      const bool R3 = (abc <= SEPS) && !R1 && !R2;
      const bool R4 = (aca <= SEPS) && !R1 && !R2 && !R3;
      const bool R5 = !(R1 || R2 || R3 || R4);

      // exp(-i x) = cos x - i sin x  (needed for every branch's formula)
      const float csa = __cosf(xa), sna = __sinf(xa);
      const float csb = __cosf(xb), snb = __sinf(xb);
      const float csc = __cosf(xc), snc = __sinf(xc);

      // Generic branch e5: share one reciprocal across the three weights:
      //   1/(sab*sca) = sbc * inv,  1/(sbc*sab) = sca * inv,  1/(sca*sbc) = sab * inv
      // with inv = 1/(sab*sbc*sca).
      const float sab = R5 ? xiab : 1.0f;
      const float sbc = R5 ? xibc : 1.0f;
      const float sca = R5 ? xica : 1.0f;
      const float inv = fast_rcp(sab * sbc * sca);
      const float w1 = sbc * inv;
      const float w2 = sca * inv;
      const float w3 = sab * inv;
      float re = csa * w1 + csb * w2 + csc * w3;
      float im = -(sna * w1 + snb * w2 + snc * w3);

      if (!R5) {  // rare degenerate regions
        if (R1) {
          const float mm = (xa + xb + xc) * (1.0f / 3.0f);
          re = 0.5f * __cosf(mm);
          im = -0.5f * __sinf(mm);
        } else if (R2) {
          const float m2 = (xa + xb) * 0.5f;
          const float id = fast_rcp(m2 - xc);
          const float cm = __cosf(m2), sm = __sinf(m2);
          re = (sm + (cm - csc) * id) * id;
          im = (cm + (snc - sm) * id) * id;
        } else if (R3) {
          const float m3 = (xb + xc) * 0.5f;
          const float id = fast_rcp(m3 - xa);
          const float cm = __cosf(m3), sm = __sinf(m3);
          re = (sm + (cm - csa) * id) * id;
          im = (cm + (sna - sm) * id) * id;
        } else {  // R4
          const float m4 = (xc + xa) * 0.5f;
          const float id = fast_rcp(m4 - xb);
          const float cm = __cosf(m4), sm = __sinf(m4);
          re = (sm + (cm - csb) * id) * id;
          im = (cm + (snb - sm) * id) * id;
        }
      }

      accRe[r] = fmaf(t, re, accRe[r]);
      accIm[r] = fmaf(t, im, accIm[r]);
    }
  }

  // Face reduction: sum the 16 lane-columns within each half-wave.
  // Then lane 0 holds rows M=r (grid gbase+r), lane 16 holds M=r+8.
#pragma unroll
  for (int r = 0; r < 8; ++r) {
    float sr = accRe[r];
    float si = accIm[r];
#pragma unroll
    for (int mask = 1; mask < 16; mask <<= 1) {
      sr += __shfl_xor(sr, mask, 32);
      si += __shfl_xor(si, mask, 32);
    }
    if (m == 0) {
      const int gg = gbase + r + hi * 8;
      const bool keep = fabsf(OTF[gg]) > THRESH;
      out[2 * gg + 0] = keep ? sr : 0.0f;
      out[2 * gg + 1] = keep ? si : 0.0f;
    }
  }
}

// ---------------------------------------------------------------------------
// Launch
// ---------------------------------------------------------------------------
extern "C" void kernel_launch(void* const* d_in, const int* in_sizes, int n_in,
                              void* d_out, int out_size, void* d_ws, size_t ws_size,
                              hipStream_t stream) {
  const float* Verts  = (const float*)d_in[0];  // 258 x 3
  const int*   Faces  = (const int*)  d_in[1];  // 512 x 3
  const float* coeff  = (const float*)d_in[2];  // 512
  const float* xi0    = (const float*)d_in[3];  // 32
  const float* xi1    = (const float*)d_in[4];  // 32
  const float* xi2    = (const float*)d_in[5];  // 32
  const float* OTF    = (const float*)d_in[6];  // 32768
  float*       out    = (float*)d_out;          // 32768 complex64 -> interleaved
  float*       ws     = (float*)d_ws;           // 512 + 6144 floats used

  mesh_ft_precompute<<<1, NF, 0, stream>>>(Verts, Faces, coeff, ws);

  // 2048 waves x 16 grid points = 32768; 8 waves/block -> 256 blocks of 256.
  mesh_ft_main<<<NGRID / (16 * 8), 256, 0, stream>>>(xi0, xi1, xi2, OTF, ws, out);
}